// AffinityLoss_2302102471387
// MI455X (gfx1250) — compile-verified
//
#include <hip/hip_runtime.h>

typedef float v2f __attribute__((ext_vector_type(2)));
typedef float v8f __attribute__((ext_vector_type(8)));

#define N_KP        21
#define N_FACE      6
#define N_PTS       121     // points per face (11x11 grid)
#define N_TILE      46      // 46*16 = 736 >= 726 columns
#define N_ENT       126     // 21 hand * 6 faces
#define N_CONT      10
#define POSE_STRIDE 87      // 29*3 floats per batch element

__device__ __constant__ int FACE_INDS_d[6][4] = {
    {0,1,2,3},{0,4,2,6},{0,1,4,5},{1,3,5,7},{2,3,6,7},{4,5,6,7}};

__device__ __forceinline__ unsigned long long wave_min_u64(unsigned long long v) {
    #pragma unroll
    for (int off = 16; off > 0; off >>= 1) {
        unsigned lo = (unsigned)v, hi = (unsigned)(v >> 32);
        unsigned olo = __shfl_xor(lo, off, 32);
        unsigned ohi = __shfl_xor(hi, off, 32);
        unsigned long long o = ((unsigned long long)ohi << 32) | olo;
        v = (o < v) ? o : v;
    }
    return v;
}

__device__ __forceinline__ void bilinear_point(const float* O, int face, int p,
                                               float& ox, float& oy, float& oz) {
    int ui = p / 11, vi = p - ui * 11;
    float u = (float)ui * 0.1f;
    float v = (float)vi * 0.1f;
    float w0 = v * (1.f - u), w1 = v * u, w2 = (1.f - v) * (1.f - u), w3 = u * (1.f - v);
    const int* F = FACE_INDS_d[face];
    int c0 = F[0] * 3, c1 = F[1] * 3, c2 = F[2] * 3, c3 = F[3] * 3;
    ox = w0 * O[c0 + 0] + w1 * O[c1 + 0] + w2 * O[c2 + 0] + w3 * O[c3 + 0];
    oy = w0 * O[c0 + 1] + w1 * O[c1 + 1] + w2 * O[c2 + 1] + w3 * O[c3 + 1];
    oz = w0 * O[c0 + 2] + w1 * O[c1 + 2] + w2 * O[c2 + 2] + w3 * O[c3 + 2];
}

__global__ __launch_bounds__(256)
void affinity_kernel(const float* __restrict__ poses, float* __restrict__ accum, int bs) {
    __shared__ float objS[8][24];
    __shared__ float xxS[8][32];
    __shared__ unsigned long long minTab[8][128];

    const int lane = threadIdx.x & 31;
    const int wv   = threadIdx.x >> 5;
    const int braw = blockIdx.x * 8 + wv;
    const bool active = (braw < bs);
    const int b = active ? braw : 0;
    const float* base = poses + (size_t)b * POSE_STRIDE;

    // ---- per-wave init -------------------------------------------------
    if (lane < 24) objS[wv][lane] = base[63 + lane];          // obj corners (8x3)
    {
        float x = 0.f;
        if (lane < N_KP) {
            float hx = base[lane*3+0], hy = base[lane*3+1], hz = base[lane*3+2];
            x = hx*hx + hy*hy + hz*hz + 1e-6f;    // fold the +1e-6 epsilon here
        }
        xxS[wv][lane] = x;
    }
    minTab[wv][lane      ] = ~0ull;
    minTab[wv][lane + 32 ] = ~0ull;
    minTab[wv][lane + 64 ] = ~0ull;
    minTab[wv][lane + 96 ] = ~0ull;
    __syncthreads();

    // ---- A tiles: hand points, K-padded [hx,hy,hz,1] -------------------
    const int m   = lane & 15;
    const int klo = (lane < 16) ? 0 : 2;
    v2f a0, a1;
    {
        a0.x = (klo + 0 < 3) ? base[m*3 + klo + 0] : 1.0f;
        a0.y = (klo + 1 < 3) ? base[m*3 + klo + 1] : 1.0f;
        int r1 = 16 + m;
        float v0 = 0.f, v1 = 0.f;
        if (r1 < N_KP) {
            v0 = (klo + 0 < 3) ? base[r1*3 + klo + 0] : 1.0f;
            v1 = (klo + 1 < 3) ? base[r1*3 + klo + 1] : 1.0f;
        }
        a1.x = v0; a1.y = v1;
    }

    // ---- hoist loop-invariant xx[row] (+eps) into registers ------------
    const int  rbase = (lane >= 16) ? 8 : 0;
    const bool lowHalf = (lane < 16);
    float xr0[8], xr1[5];
    #pragma unroll
    for (int vg = 0; vg < 8; ++vg) xr0[vg] = xxS[wv][rbase + vg];
    #pragma unroll
    for (int vg = 0; vg < 5; ++vg) xr1[vg] = xxS[wv][16 + vg];  // rows 16..20 (used by lanes 0-15)

    // ---- WMMA over 46 column tiles of object surface points ------------
    const float* O = objS[wv];
    for (int t = 0; t < N_TILE; ++t) {
        int j = t * 16 + m;
        int face = j / N_PTS;
        int p = j - face * N_PTS;
        float ox = 0.f, oy = 0.f, oz = 0.f, yy = 1e30f;
        if (face < N_FACE) {
            bilinear_point(O, face, p, ox, oy, oz);
            yy = ox*ox + oy*oy + oz*oz;
        }
        float bk0, bk1;
        if (klo == 0) { bk0 = -2.f * ox; bk1 = -2.f * oy; }
        else          { bk0 = -2.f * oz; bk1 = yy;        }
        v2f bv; bv.x = bk0; bv.y = bk1;

        v8f c0 = {0.f,0.f,0.f,0.f,0.f,0.f,0.f,0.f};
        v8f c1 = {0.f,0.f,0.f,0.f,0.f,0.f,0.f,0.f};
        c0 = __builtin_amdgcn_wmma_f32_16x16x4_f32(false, a0, false, bv, (short)0, c0, false, false);
        c1 = __builtin_amdgcn_wmma_f32_16x16x4_f32(false, a1, false, bv, (short)0, c1, false, false);

        if (face < N_FACE) {
            #pragma unroll
            for (int vg = 0; vg < 8; ++vg) {
                {   // tile0: rows 0..15, all valid
                    int row = rbase + vg;
                    float d2 = c0[vg] + xr0[vg];
                    unsigned long long key =
                        ((unsigned long long)__float_as_uint(d2) << 32) | (unsigned)p;
                    atomicMin(&minTab[wv][row * N_FACE + face], key);
                }
                if (vg < 5) {            // tile1: only rows 16..20 are real hand points
                    if (lowHalf) {       // lanes 0-15 hold M=16+vg; lanes 16-31 map to rows >=24
                        float d2 = c1[vg] + xr1[vg];
                        unsigned long long key =
                            ((unsigned long long)__float_as_uint(d2) << 32) | (unsigned)p;
                        atomicMin(&minTab[wv][(16 + vg) * N_FACE + face], key);
                    }
                }
            }
        }
    }
    __syncthreads();

    // ---- top-10 selection over 126 (hand,face) entries -----------------
    unsigned usedMask = 0;
    unsigned long long myKey[4];
    #pragma unroll
    for (int i = 0; i < 4; ++i) {
        int e = lane + 32 * i;
        unsigned long long v = minTab[wv][e];
        unsigned d2bits = (unsigned)(v >> 32);
        myKey[i] = (e < N_ENT) ? (((unsigned long long)d2bits << 32) | (unsigned)e) : ~0ull;
    }
    int selE[N_CONT];
    unsigned selBits[N_CONT];
    for (int k = 0; k < N_CONT; ++k) {
        unsigned long long best = ~0ull;
        #pragma unroll
        for (int i = 0; i < 4; ++i)
            if (!(usedMask & (1u << i)) && myKey[i] < best) best = myKey[i];
        best = wave_min_u64(best);
        int e = (int)(best & 0xffffffffu);
        selE[k] = e;
        selBits[k] = (unsigned)(best >> 32);
        if ((e & 31) == lane) usedMask |= 1u << (e >> 5);
    }

    // ---- contact geometry (lanes 0..9, one contact each) ---------------
    float sx = 0.f, sy = 0.f, sz = 0.f, sm = 0.f;
    if (lane < N_CONT) {
        int e = selE[lane];
        int face = e % N_FACE;
        int p = (int)(minTab[wv][e] & 0xffffffffu);
        float dist = sqrtf(__uint_as_float(selBits[lane]));

        float cx, cy, cz;
        bilinear_point(O, face, p, cx, cy, cz);

        float p1x = 0.25f*(O[0]+O[3]+O[6]+O[9]);
        float p1y = 0.25f*(O[1]+O[4]+O[7]+O[10]);
        float p1z = 0.25f*(O[2]+O[5]+O[8]+O[11]);
        float p2x = 0.25f*(O[12]+O[15]+O[18]+O[21]);
        float p2y = 0.25f*(O[13]+O[16]+O[19]+O[22]);
        float p2z = 0.25f*(O[14]+O[17]+O[20]+O[23]);

        float l1 = 0.f, l2 = 0.f;
        #pragma unroll
        for (int i = 0; i < 4; ++i) {
            int i2 = (i + 1) & 3;
            float dx = O[i*3+0]-O[i2*3+0], dy = O[i*3+1]-O[i2*3+1], dz = O[i*3+2]-O[i2*3+2];
            l1 += sqrtf(dx*dx + dy*dy + dz*dz);
            float ex = O[(i+4)*3+0]-O[(i2+4)*3+0];
            float ey = O[(i+4)*3+1]-O[(i2+4)*3+1];
            float ez = O[(i+4)*3+2]-O[(i2+4)*3+2];
            l2 += sqrtf(ex*ex + ey*ey + ez*ez);
        }
        l1 *= 0.25f; l2 *= 0.25f;
        float length = 0.5f * (l1 + l2);
        float mk = (dist < length * 0.2f) ? 1.0f : 0.0f;

        float dvx = p2x - p1x, dvy = p2y - p1y, dvz = p2z - p1z;
        float dvn = sqrtf(dvx*dvx + dvy*dvy + dvz*dvz);
        float inv = 1.0f / (dvn + 1e-5f);
        float ndx = dvx * inv, ndy = dvy * inv, ndz = dvz * inv;

        float vx = cx - p1x, vy = cy - p1y, vz = cz - p1z;
        float tpar = (dvx*vx + dvy*vy + dvz*vz) * inv;
        float rx = p1x + ndx * tpar, ry = p1y + ndy * tpar, rz = p1z + ndz * tpar;
        float nx = cx - rx, ny = cy - ry, nz = cz - rz;
        float nlen = sqrtf(nx*nx + ny*ny + nz*nz);
        float ninv = 1.0f / (nlen + 1e-5f);
        nx *= ninv; ny *= ninv; nz *= ninv;

        sx = mk * nx; sy = mk * ny; sz = mk * nz; sm = mk;
    }

    // ---- wave reduce and global accumulate -----------------------------
    #pragma unroll
    for (int off = 16; off > 0; off >>= 1) {
        sx += __shfl_xor(sx, off, 32);
        sy += __shfl_xor(sy, off, 32);
        sz += __shfl_xor(sz, off, 32);
        sm += __shfl_xor(sm, off, 32);
    }
    if (lane == 0 && active) {
        atomicAdd(&accum[0], sx*sx + sy*sy + sz*sz);   // sum cos_sim * mask
        atomicAdd(&accum[1], sm*sm);                   // sum mask
    }
}

__global__ void init_accum_kernel(float* accum) {
    if (threadIdx.x < 2) accum[threadIdx.x] = 0.f;
}

__global__ void finalize_kernel(const float* __restrict__ accum, float* __restrict__ out) {
    out[0] = accum[0] / (accum[1] + 1.0f);
}

extern "C" void kernel_launch(void* const* d_in, const int* in_sizes, int n_in,
                              void* d_out, int out_size, void* d_ws, size_t ws_size,
                              hipStream_t stream) {
    (void)n_in; (void)out_size; (void)ws_size;
    const float* poses = (const float*)d_in[0];
    float* out = (float*)d_out;
    float* accum = (float*)d_ws;

    int total = in_sizes[0];
    int bs = total / POSE_STRIDE;          // 8192 for the reference setup
    int nblocks = (bs + 7) / 8;            // 8 waves (batch elems) per 256-thread block

    init_accum_kernel<<<1, 64, 0, stream>>>(accum);
    affinity_kernel<<<nblocks, 256, 0, stream>>>(poses, accum, bs);
    finalize_kernel<<<1, 1, 0, stream>>>(accum, out);
}